// NMS_34299608826377
// MI455X (gfx1250) — compile-verified
//
#include <hip/hip_runtime.h>
#include <stdint.h>

// ---------------- problem constants ----------------
#define B     16
#define N     25200
#define C     85
#define KTOP  1000
#define NPAD  32768          // 2^15 >= N, for bitonic sort
#define KPAD  1024           // padded K (32 mask words * 32 bits), == blockDim of mask kernel
#define CHUNK 2048           // LDS bitonic chunk
#define CONF  0.25f
#define IOUT  0.45f
#define MAXWH 4096.0f

typedef float v2f __attribute__((ext_vector_type(2)));
typedef float v8f __attribute__((ext_vector_type(8)));
typedef int   v4i __attribute__((ext_vector_type(4)));

typedef __attribute__((address_space(1))) v4i* gbl_v4i_p;   // global
typedef __attribute__((address_space(3))) v4i* lds_v4i_p;   // LDS

#if __has_builtin(__builtin_amdgcn_global_load_async_to_lds_b128)
#define HAVE_ASYNC_LDS 1
#endif

__device__ __forceinline__ void wait_asynccnt0() {
#if __has_builtin(__builtin_amdgcn_s_wait_asynccnt)
  __builtin_amdgcn_s_wait_asynccnt(0);
#else
  asm volatile("s_wait_asynccnt 0x0" ::: "memory");
#endif
}

// ---------------------------------------------------------------
// Kernel 1: per-candidate score, class argmax, sortable 64-bit key
// key = score_bits << 32 | (0xFFFFFFFF - n)  (stable ties: low n first)
// ---------------------------------------------------------------
__global__ void score_key_kernel(const float* __restrict__ x,
                                 uint64_t* __restrict__ keys,
                                 int* __restrict__ clsb) {
  int t = blockIdx.x * blockDim.x + threadIdx.x;
  if (t >= B * NPAD) return;
  int b = t / NPAD, n = t % NPAD;
  uint64_t key = 0;
  if (n < N) {
    const float* p = x + ((size_t)b * N + n) * C;
    float obj = p[4];
    float best = -1.0f;
    int jb = 0;
    #pragma unroll 8
    for (int c = 0; c < C - 5; ++c) {
      float v = obj * p[5 + c];
      if (v > best) { best = v; jb = c; }   // strict > == argmax-first
    }
    clsb[b * N + n] = jb;
    float score = ((obj > CONF) && (best > CONF)) ? best : 0.0f;
    unsigned sb = __float_as_uint(score);   // score >= 0 -> bit-order == value-order
    key = ((uint64_t)sb << 32) | (uint32_t)(0xFFFFFFFFu - (unsigned)n);
  }
  keys[(size_t)b * NPAD + n] = key;
}

// ---------------------------------------------------------------
// Kernel 2a: LDS chunk sort — all bitonic stages k = 2..CHUNK.
// One 1024-thread workgroup sorts a 2048-key chunk in 16KB LDS.
// Direction from in-batch index so the network is globally consistent.
// ---------------------------------------------------------------
__global__ __launch_bounds__(1024) void bitonic_local_sort(uint64_t* __restrict__ keys) {
  __shared__ uint64_t s[CHUNK];
  size_t base = (size_t)blockIdx.x * CHUNK;
  s[threadIdx.x]        = keys[base + threadIdx.x];
  s[threadIdx.x + 1024] = keys[base + threadIdx.x + 1024];
  __syncthreads();
  for (int k = 2; k <= CHUNK; k <<= 1) {
    for (int j = k >> 1; j > 0; j >>= 1) {
      int tid = threadIdx.x;
      int i = ((tid & ~(j - 1)) << 1) | (tid & (j - 1));   // (i & j) == 0
      unsigned inb = ((unsigned)(base + (size_t)i)) & (NPAD - 1);
      bool desc = ((inb & (unsigned)k) == 0);
      uint64_t a = s[i], c = s[i + j];
      bool sw = desc ? (a < c) : (a > c);
      if (sw) { s[i] = c; s[i + j] = a; }
      __syncthreads();
    }
  }
  keys[base + threadIdx.x]        = s[threadIdx.x];
  keys[base + threadIdx.x + 1024] = s[threadIdx.x + 1024];
}

// ---------------------------------------------------------------
// Kernel 2b: one global cross-chunk compare-exchange pass (j >= CHUNK)
// ---------------------------------------------------------------
__global__ void bitonic_pass(uint64_t* __restrict__ keys, int k, int j) {
  int t = blockIdx.x * blockDim.x + threadIdx.x;
  if (t >= B * NPAD) return;
  int b = t / NPAD, i = t % NPAD;
  int ixj = i ^ j;
  if (ixj <= i) return;
  uint64_t* kb = keys + (size_t)b * NPAD;
  uint64_t a = kb[i], c = kb[ixj];
  bool desc = ((i & k) == 0);
  bool sw = desc ? (a < c) : (a > c);
  if (sw) { kb[i] = c; kb[ixj] = a; }
}

// ---------------------------------------------------------------
// Kernel 2c: LDS chunk merge — stages j = 1024..1 for a fixed k > CHUNK
// ---------------------------------------------------------------
__global__ __launch_bounds__(1024) void bitonic_local_merge(uint64_t* __restrict__ keys, int k) {
  __shared__ uint64_t s[CHUNK];
  size_t base = (size_t)blockIdx.x * CHUNK;
  s[threadIdx.x]        = keys[base + threadIdx.x];
  s[threadIdx.x + 1024] = keys[base + threadIdx.x + 1024];
  __syncthreads();
  for (int j = CHUNK >> 1; j > 0; j >>= 1) {
    int tid = threadIdx.x;
    int i = ((tid & ~(j - 1)) << 1) | (tid & (j - 1));
    unsigned inb = ((unsigned)(base + (size_t)i)) & (NPAD - 1);
    bool desc = ((inb & (unsigned)k) == 0);
    uint64_t a = s[i], c = s[i + j];
    bool sw = desc ? (a < c) : (a > c);
    if (sw) { s[i] = c; s[i + j] = a; }
    __syncthreads();
  }
  keys[base + threadIdx.x]        = s[threadIdx.x];
  keys[base + threadIdx.x + 1024] = s[threadIdx.x + 1024];
}

// ---------------------------------------------------------------
// Kernel 3: gather top-K: xyxy boxes -> out, offset boxes + score -> ws
// ---------------------------------------------------------------
__global__ void gather_kernel(const float* __restrict__ x,
                              const uint64_t* __restrict__ keys,
                              const int* __restrict__ clsb,
                              float* __restrict__ out,
                              float* __restrict__ selbox,
                              float* __restrict__ selscore) {
  int t = blockIdx.x * blockDim.x + threadIdx.x;
  if (t >= B * KTOP) return;
  int b = t / KTOP, kk = t % KTOP;
  uint64_t key = keys[(size_t)b * NPAD + kk];
  unsigned idx = 0xFFFFFFFFu - (unsigned)(key & 0xFFFFFFFFu);
  if (idx >= N) idx = 0;                         // padding safety (never in top-K)
  float score = __uint_as_float((unsigned)(key >> 32));
  const float* p = x + ((size_t)b * N + idx) * C;
  float cx = p[0], cy = p[1], w = p[2], h = p[3];
  float x1 = cx - 0.5f * w, y1 = cy - 0.5f * h;
  float x2 = cx + 0.5f * w, y2 = cy + 0.5f * h;
  int cls = clsb[b * N + idx];
  float off = (float)cls * MAXWH;
  float* o = out + (size_t)t * 6;
  o[0] = x1; o[1] = y1; o[2] = x2; o[3] = y2; o[5] = (float)cls;
  float* sb = selbox + (size_t)t * 4;
  sb[0] = x1 + off; sb[1] = y1 + off; sb[2] = x2 + off; sb[3] = y2 + off;
  selscore[t] = score;
}

// ---------------------------------------------------------------
// Kernel 4: suppression bit-matrix. One workgroup (32 waves) per batch.
// Each wave computes a 16-row x 32-col tile of the KxK IoU predicate.
// The pairwise (area_i + area_j) term is a rank-2 GEMM computed with
// V_WMMA_F32_16X16X4_F32: A=[area_row | 1 | 0 | 0], B=[1 ; area_col ; 0 ; 0].
// D-matrix layout: VGPR v, lanes 0-15 -> row v, lanes 16-31 -> row v+8,
// col = lane%16. Bits packed per row with wave32 ballot. The IoU threshold
// is evaluated division-free: iou > T  <=>  inter > T*(areaSum - inter + eps).
// Boxes are staged into LDS via CDNA5 async global->LDS B128 copies.
// ---------------------------------------------------------------
__global__ __launch_bounds__(1024) void mask_kernel(const float* __restrict__ selbox,
                                                    unsigned* __restrict__ mask) {
  __shared__ float4 s_box[KPAD];
  __shared__ float  s_area[KPAD];
  int b = blockIdx.x;
  int tid = threadIdx.x;

#ifdef HAVE_ASYNC_LDS
  if (tid < KTOP) {
    const float* sp = selbox + ((size_t)b * KTOP + tid) * 4;
    __builtin_amdgcn_global_load_async_to_lds_b128(
        (gbl_v4i_p)sp,
        (lds_v4i_p)&s_box[tid],
        0, 0);
  }
  wait_asynccnt0();
  __syncthreads();
  if (tid >= KTOP) s_box[tid] = make_float4(-1e8f, -1e8f, -1e8f, -1e8f);
  {
    float4 bx = s_box[tid];
    s_area[tid] = (bx.z - bx.x) * (bx.w - bx.y);
  }
  __syncthreads();
#else
  {
    float4 bx;
    if (tid < KTOP) {
      const float* sp = selbox + ((size_t)b * KTOP + tid) * 4;
      bx = make_float4(sp[0], sp[1], sp[2], sp[3]);
    } else {
      bx = make_float4(-1e8f, -1e8f, -1e8f, -1e8f);  // degenerate: iou == 0
    }
    s_box[tid] = bx;
    s_area[tid] = (bx.z - bx.x) * (bx.w - bx.y);
  }
  __syncthreads();
#endif

  int wave = tid >> 5;
  int lane = tid & 31;

  // 63 row-tiles (rows 0..1007 cover K=1000) x 32 word-tiles = 2016 tiles
  for (int tile = wave; tile < 63 * 32; tile += 32) {
    int rt = tile >> 5;          // row tile 0..62
    int wt = tile & 31;          // 32-bit mask word index 0..31
    int rbase = rt * 16;

    // A matrix (16x4 f32): lanes 0-15: VGPR0=K0=area_row[M], VGPR1=K1=1
    //                      lanes 16-31: K2=K3=0
    v2f a;
    if (lane < 16) { a.x = s_area[rbase + lane]; a.y = 1.0f; }
    else           { a.x = 0.0f;                 a.y = 0.0f; }

    unsigned lo[8], hi[8];
    #pragma unroll
    for (int half = 0; half < 2; ++half) {
      int cbase = wt * 32 + half * 16;
      // B matrix (4x16 f32): VGPR0: lanes 0-15 = K0 row (ones),
      //                              lanes 16-31 = K1 row (area_col[N])
      v2f bv;
      bv.x = (lane < 16) ? 1.0f : s_area[cbase + (lane - 16)];
      bv.y = 0.0f;
      v8f cz = {0.f, 0.f, 0.f, 0.f, 0.f, 0.f, 0.f, 0.f};
      v8f d = __builtin_amdgcn_wmma_f32_16x16x4_f32(
          false, a, false, bv, (short)0, cz, false, false);

      float4 cb = s_box[cbase + (lane & 15)];
      int roff = rbase + ((lane >= 16) ? 8 : 0);
      #pragma unroll
      for (int v = 0; v < 8; ++v) {
        float4 rb = s_box[roff + v];
        float ix1 = fmaxf(rb.x, cb.x), iy1 = fmaxf(rb.y, cb.y);
        float ix2 = fminf(rb.z, cb.z), iy2 = fminf(rb.w, cb.w);
        float iw = fmaxf(ix2 - ix1, 0.0f), ih = fmaxf(iy2 - iy1, 0.0f);
        float inter = iw * ih;
        // d[v] = area_i + area_j (from WMMA); union + eps > 0 always
        float denom = d[v] - inter + 1e-7f;
        unsigned bm = (unsigned)__ballot(inter > IOUT * denom);
        if (half == 0) {
          lo[v] = bm & 0xFFFFu;
          hi[v] = bm >> 16;
        } else {
          unsigned w0 = lo[v] | ((bm & 0xFFFFu) << 16);   // row rbase+v
          unsigned w1 = hi[v] | ((bm >> 16) << 16);       // row rbase+v+8
          if (lane == 0) {
            int r0 = rbase + v, r1 = rbase + v + 8;
            if (r0 < KTOP) mask[((size_t)b * KTOP + r0) * 32 + wt] = w0;
            if (r1 < KTOP) mask[((size_t)b * KTOP + r1) * 32 + wt] = w1;
          }
        }
      }
    }
  }
}

// ---------------------------------------------------------------
// Kernel 5: sequential greedy scan. One wave32 per batch; lane l owns
// keep bits [32l, 32l+31]. keep[i] broadcast via shuffle.
// ---------------------------------------------------------------
__global__ void scan_kernel(const unsigned* __restrict__ mask,
                            unsigned* __restrict__ keepbuf) {
  int b = blockIdx.x;
  int lane = threadIdx.x;                       // 32 threads
  unsigned keepw = (lane < 31) ? 0xFFFFFFFFu : 0xFFu;   // bits >= K cleared
  for (int i = 0; i < KTOP; ++i) {
    int word = i >> 5, bit = i & 31;
    unsigned kw = __shfl(keepw, word, 32);
    if ((kw >> bit) & 1u) {
      unsigned m = mask[((size_t)b * KTOP + i) * 32 + lane];
      if (lane < word)       m = 0;                         // only suppress j > i
      else if (lane == word) m &= ~((2u << bit) - 1u);      // clear bits <= i
      keepw &= ~m;
    }
  }
  keepbuf[b * 32 + lane] = keepw;
}

// ---------------------------------------------------------------
// Kernel 6: final score column
// ---------------------------------------------------------------
__global__ void finalize_kernel(const float* __restrict__ selscore,
                                const unsigned* __restrict__ keepbuf,
                                float* __restrict__ out) {
  int t = blockIdx.x * blockDim.x + threadIdx.x;
  if (t >= B * KTOP) return;
  int b = t / KTOP, k = t % KTOP;
  float sc = selscore[t];
  unsigned kept = (keepbuf[b * 32 + (k >> 5)] >> (k & 31)) & 1u;
  out[(size_t)t * 6 + 4] = (kept && (sc > CONF)) ? sc : 0.0f;
}

// ---------------------------------------------------------------
extern "C" void kernel_launch(void* const* d_in, const int* in_sizes, int n_in,
                              void* d_out, int out_size, void* d_ws, size_t ws_size,
                              hipStream_t stream) {
  const float* x = (const float*)d_in[0];
  float* out = (float*)d_out;

  char* ws = (char*)d_ws;                         // ~7.8 MB total
  uint64_t* keys   = (uint64_t*)ws;  ws += (size_t)B * NPAD * 8;        // 4.0 MB
  int*      clsb   = (int*)ws;       ws += (size_t)B * N * 4;           // 1.6 MB
  float*    selbox = (float*)ws;     ws += (size_t)B * KTOP * 4 * 4;    // 256 KB
  float*    selsc  = (float*)ws;     ws += (size_t)B * KTOP * 4;        // 64 KB
  unsigned* mask   = (unsigned*)ws;  ws += (size_t)B * KTOP * 32 * 4;   // 2.0 MB
  unsigned* keepb  = (unsigned*)ws;                                      // 2 KB

  int tot = B * NPAD;
  score_key_kernel<<<(tot + 255) / 256, 256, 0, stream>>>(x, keys, clsb);

  // hierarchical descending bitonic sort per batch (keys stay in 192MB L2):
  // 1 LDS chunk-sort + 10 global passes + 4 LDS merges  (was 120 global passes)
  int nchunks = (B * NPAD) / CHUNK;               // 256
  bitonic_local_sort<<<nchunks, 1024, 0, stream>>>(keys);
  for (int k = 2 * CHUNK; k <= NPAD; k <<= 1) {
    for (int j = k >> 1; j >= CHUNK; j >>= 1)
      bitonic_pass<<<(tot + 255) / 256, 256, 0, stream>>>(keys, k, j);
    bitonic_local_merge<<<nchunks, 1024, 0, stream>>>(keys, k);
  }

  int bk = B * KTOP;
  gather_kernel<<<(bk + 255) / 256, 256, 0, stream>>>(x, keys, clsb, out, selbox, selsc);
  mask_kernel<<<B, 1024, 0, stream>>>(selbox, mask);
  scan_kernel<<<B, 32, 0, stream>>>(mask, keepb);
  finalize_kernel<<<(bk + 255) / 256, 256, 0, stream>>>(selsc, keepb, out);
}